// SpatiotemporalUncertaintyLoss_16140487098675
// MI455X (gfx1250) — compile-verified
//
#include <hip/hip_runtime.h>
#include <hip/hip_bf16.h>

// ---------------------------------------------------------------------------
// SpatiotemporalUncertaintyLoss for MI455X (gfx1250, wave32).
// Lidar: atomic scatter segment-sums + pointwise finalize (memory bound).
// Radar: pred-vs-gt nearest neighbor via V_WMMA_F32_16X16X4_F32 tiles
//        (d2 = a2[m] + b2[n] - 2*pred@gt^T, min over n), gt staged in LDS.
// ---------------------------------------------------------------------------

typedef __attribute__((ext_vector_type(2))) float v2f;
typedef __attribute__((ext_vector_type(8))) float v8f;

// ---- constants (float-accurate precomputed from the reference) ------------
#define L_MIN_C  (-11.618285313961388f)   // 2*ln(0.03/10 + 1e-9)
#define L_MAX_C  (-5.991464507107982f)    // 2*ln(0.5/10  + 1e-9)
#define R_MIN_C  (-7.824045910856292f)    // 2*ln(0.1/5   + 1e-9)
#define R_MAX_C  (2.0e-9f)                // 2*ln(5.0/5   + 1e-9)
#define GHOST_C  (0.0036f)                // (0.6/10)^2

__device__ __forceinline__ void atomAddF(float* p, float v) {
  unsafeAtomicAdd(p, v);   // global_atomic_add_f32 (no return)
}

__device__ __forceinline__ float blockReduceSum256(float v) {
  __shared__ float red[256];
  red[threadIdx.x] = v;
  __syncthreads();
  for (int s = 128; s > 0; s >>= 1) {
    if ((int)threadIdx.x < s) red[threadIdx.x] += red[threadIdx.x + s];
    __syncthreads();
  }
  return red[0];
}

// ---------------------------------------------------------------------------
// 0) zero workspace
// ---------------------------------------------------------------------------
__global__ void zero_kernel(float* __restrict__ p, long long n) {
  long long i = (long long)blockIdx.x * blockDim.x + threadIdx.x;
  long long stride = (long long)gridDim.x * blockDim.x;
  for (; i < n; i += stride) p[i] = 0.0f;
}

// ---------------------------------------------------------------------------
// 1) lidar edge scatter: cnt/sum_pos/sum_int by dst
// ---------------------------------------------------------------------------
__global__ void lidar_scatter_kernel(const int* __restrict__ lsrc,
                                     const int* __restrict__ ldst,
                                     const float* __restrict__ pos,
                                     const float* __restrict__ x,
                                     float* __restrict__ cnt,
                                     float* __restrict__ spos,
                                     float* __restrict__ sint,
                                     int e) {
  int i = blockIdx.x * blockDim.x + threadIdx.x;
  if (i >= e) return;
  int s = lsrc[i];
  int d = ldst[i];
  float px = pos[3 * s + 0], py = pos[3 * s + 1], pz = pos[3 * s + 2];
  float it = x[4 * s + 2];
  atomAddF(cnt + d, 1.0f);
  atomAddF(spos + 3 * d + 0, px);
  atomAddF(spos + 3 * d + 1, py);
  atomAddF(spos + 3 * d + 2, pz);
  atomAddF(sint + d, it);
}

// ---------------------------------------------------------------------------
// 2) lidar finalize: sum of 0.5*exp(-lv)*(res_pos+res_int) + 0.5*lv
// ---------------------------------------------------------------------------
__global__ void lidar_finalize_kernel(const float* __restrict__ logvar,
                                      const float* __restrict__ pos,
                                      const float* __restrict__ x,
                                      const float* __restrict__ cnt,
                                      const float* __restrict__ spos,
                                      const float* __restrict__ sint,
                                      float* __restrict__ acc, int n) {
  int i = blockIdx.x * blockDim.x + threadIdx.x;
  float contrib = 0.0f;
  if (i < n) {
    float lv = fminf(fmaxf(logvar[i], L_MIN_C), L_MAX_C);
    float c = cnt[i];
    float denom = fmaxf(c, 1.0f);
    float inv = 1.0f / denom;
    float mx = spos[3 * i + 0] * inv;
    float my = spos[3 * i + 1] * inv;
    float mz = spos[3 * i + 2] * inv;
    float mi = sint[i] * inv;
    float dx = pos[3 * i + 0] - mx;
    float dy = pos[3 * i + 1] - my;
    float dz = pos[3 * i + 2] - mz;
    float di = x[4 * i + 2] - mi;
    float res_pos = dx * dx + dy * dy + dz * dz;
    float res_int = di * di;
    float prec = __expf(-lv);
    contrib = 0.5f * prec * (res_pos + res_int) + 0.5f * lv;
  }
  float bs = blockReduceSum256(contrib);
  if (threadIdx.x == 0) atomAddF(acc, bs);
}

// ---------------------------------------------------------------------------
// 3) radar pred:  pred = pos[s] + |x[s,2]| * unit(pos[d]-pos[s]) * dt[s]
// ---------------------------------------------------------------------------
__global__ void radar_pred_kernel(const float* __restrict__ pos,
                                  const float* __restrict__ x,
                                  const int* __restrict__ tsrc,
                                  const int* __restrict__ tdst,
                                  const int* __restrict__ batch,
                                  const float* __restrict__ dt_sec,
                                  float* __restrict__ pred,
                                  float* __restrict__ a2, int n) {
  int i = blockIdx.x * blockDim.x + threadIdx.x;
  if (i >= n) return;
  int s = tsrc[i];
  int d = tdst[i];
  float sx = pos[3 * s + 0], sy = pos[3 * s + 1], sz = pos[3 * s + 2];
  float mx = pos[3 * d + 0] - sx;
  float my = pos[3 * d + 1] - sy;
  float mz = pos[3 * d + 2] - sz;
  float nrm = sqrtf(mx * mx + my * my + mz * mz) + 1e-9f;
  float inv = 1.0f / nrm;
  float speed = fabsf(x[4 * s + 2]);
  float dt = fmaxf(dt_sec[batch[s]], 0.01f);
  float f = speed * dt * inv;
  float px = sx + f * mx;
  float py = sy + f * my;
  float pz = sz + f * mz;
  pred[3 * i + 0] = px;
  pred[3 * i + 1] = py;
  pred[3 * i + 2] = pz;
  a2[i] = px * px + py * py + pz * pz;
}

// ---------------------------------------------------------------------------
// 4) radar NN search via WMMA f32 16x16x4.
//    One wave owns 16 pred rows; loops over gt in 16-column tiles.
//    D fragment (8 VGPRs): lanes 0-15 -> M=v, lanes 16-31 -> M=v+8, N=lane&15.
//    A (16x4): lane<16 holds {x,y}=K0,K1 of row lane; lane>=16 holds {z,0}=K2,K3.
//    B (4x16): mirrored (lane<16: K0,K1 of col lane; lane>=16: K2,K3).
// ---------------------------------------------------------------------------
#define MAX_GT 2048
__global__ void radar_nn_wmma_kernel(const float* __restrict__ pred,
                                     const float* __restrict__ a2,
                                     const float* __restrict__ gt,
                                     const int* __restrict__ tsrc,
                                     float* __restrict__ physics,
                                     int n, int m_gt) {
  __shared__ float sgt[MAX_GT * 3];
  int gtf = m_gt * 3;
  if (gtf > MAX_GT * 3) gtf = MAX_GT * 3;
  for (int i = threadIdx.x; i < gtf; i += blockDim.x) sgt[i] = gt[i];
  __syncthreads();

  int lane = threadIdx.x & 31;
  int waveInBlk = threadIdx.x >> 5;
  int wave = blockIdx.x * (blockDim.x >> 5) + waveInBlk;
  int base = wave * 16;                 // first pred row of this wave's tile
  if (base >= n) return;                // wave-uniform: EXEC stays all-ones

  int msub = lane & 15;
  int hi = lane >> 4;

  int node = base + msub;
  int nodeL = node < n ? node : (n - 1);   // clamp OOB loads (values unused)
  v2f afrag;
  if (hi == 0) {
    afrag.x = pred[3 * nodeL + 0];
    afrag.y = pred[3 * nodeL + 1];
  } else {
    afrag.x = pred[3 * nodeL + 2];
    afrag.y = 0.0f;
  }

  float runm[8];
#pragma unroll
  for (int v = 0; v < 8; ++v) runm[v] = 3.0e38f;

  for (int t = 0; t < m_gt; t += 16) {
    int col = t + msub;
    float gx = sgt[3 * col + 0];
    float gy = sgt[3 * col + 1];
    float gz = sgt[3 * col + 2];
    float b2 = gx * gx + gy * gy + gz * gz;
    v2f bfrag;
    if (hi == 0) {
      bfrag.x = gx;
      bfrag.y = gy;
    } else {
      bfrag.x = gz;
      bfrag.y = 0.0f;
    }
    v8f c = {};
    c = __builtin_amdgcn_wmma_f32_16x16x4_f32(
        /*neg_a=*/false, afrag, /*neg_b=*/false, bfrag,
        /*c_mod=*/(short)0, c, /*reuse_a=*/false, /*reuse_b=*/false);
#pragma unroll
    for (int v = 0; v < 8; ++v) {
      float d = b2 - 2.0f * c[v];       // a2[m] added after the reduction
      runm[v] = fminf(runm[v], d);
    }
  }

  // min across the 16 lanes of each half (masks 1,2,4,8 stay within a half)
#pragma unroll
  for (int v = 0; v < 8; ++v) {
    float r = runm[v];
    r = fminf(r, __shfl_xor(r, 1, 32));
    r = fminf(r, __shfl_xor(r, 2, 32));
    r = fminf(r, __shfl_xor(r, 4, 32));
    r = fminf(r, __shfl_xor(r, 8, 32));
    runm[v] = r;
  }

  if (msub == 0) {                      // lane 0 -> rows 0..7, lane 16 -> 8..15
#pragma unroll
    for (int v = 0; v < 8; ++v) {
      int m = hi * 8 + v;
      int i = base + m;
      if (i < n) {
        float d2 = fmaxf(a2[i] + runm[v], 0.0f);
        physics[tsrc[i]] = d2;
      }
    }
  }
}

// ---------------------------------------------------------------------------
// 5) radar->lidar edge scatter: sum_d / cnt by rl_src
// ---------------------------------------------------------------------------
__global__ void radar_rl_scatter_kernel(const int* __restrict__ rlsrc,
                                        const int* __restrict__ rldst,
                                        const float* __restrict__ rpos,
                                        const float* __restrict__ lpos,
                                        float* __restrict__ sumd,
                                        float* __restrict__ cntr, int e) {
  int i = blockIdx.x * blockDim.x + threadIdx.x;
  if (i >= e) return;
  int rs = rlsrc[i];
  int rd = rldst[i];
  float dx = rpos[3 * rs + 0] - lpos[3 * rd + 0];
  float dy = rpos[3 * rs + 1] - lpos[3 * rd + 1];
  float dz = rpos[3 * rs + 2] - lpos[3 * rd + 2];
  float d2 = dx * dx + dy * dy + dz * dz;
  atomAddF(sumd + rs, d2);
  atomAddF(cntr + rs, 1.0f);
}

// ---------------------------------------------------------------------------
// 6) radar finalize: (physics + spatial)/denom + 0.5*lv
// ---------------------------------------------------------------------------
__global__ void radar_finalize_kernel(const float* __restrict__ logvar,
                                      const int* __restrict__ batch,
                                      const float* __restrict__ dt_sec,
                                      const float* __restrict__ physics,
                                      const float* __restrict__ sumd,
                                      const float* __restrict__ cntr,
                                      float* __restrict__ acc, int n) {
  int i = blockIdx.x * blockDim.x + threadIdx.x;
  float contrib = 0.0f;
  if (i < n) {
    float lv = fminf(fmaxf(logvar[i], R_MIN_C), R_MAX_C);
    float dt = fmaxf(dt_sec[batch[i]], 0.01f);
    float c = cntr[i];
    float cm = fmaxf(c, 1.0f);
    float spatial = (c > 0.0f) ? (sumd[i] / (cm * cm)) : GHOST_C;
    float denom = 2.0f * __expf(lv) * dt * dt + 1e-9f;
    contrib = (physics[i] + spatial) / denom + 0.5f * lv;
  }
  float bs = blockReduceSum256(contrib);
  if (threadIdx.x == 0) atomAddF(acc, bs);
}

// ---------------------------------------------------------------------------
// 7) combine
// ---------------------------------------------------------------------------
__global__ void final_kernel(const float* __restrict__ acc, float* __restrict__ out,
                             int n_l, int n_r) {
  if (threadIdx.x == 0 && blockIdx.x == 0) {
    out[0] = acc[0] / (float)n_l + acc[1] / (float)n_r;
  }
}

// ---------------------------------------------------------------------------
extern "C" void kernel_launch(void* const* d_in, const int* in_sizes, int n_in,
                              void* d_out, int out_size, void* d_ws, size_t ws_size,
                              hipStream_t stream) {
  const float* lidar_logvar = (const float*)d_in[0];
  const float* lidar_pos    = (const float*)d_in[1];
  const float* lidar_x      = (const float*)d_in[2];
  const float* r_logvar[2]  = {(const float*)d_in[3], (const float*)d_in[6]};
  const float* r_pos[2]     = {(const float*)d_in[4], (const float*)d_in[7]};
  const float* r_x[2]       = {(const float*)d_in[5], (const float*)d_in[8]};
  const float* gt_pos       = (const float*)d_in[9];
  const float* dt_sec       = (const float*)d_in[10];
  const int*   l_src        = (const int*)d_in[11];
  const int*   l_dst        = (const int*)d_in[12];
  const int*   r_batch[2]   = {(const int*)d_in[13], (const int*)d_in[18]};
  const int*   r_tsrc[2]    = {(const int*)d_in[14], (const int*)d_in[19]};
  const int*   r_tdst[2]    = {(const int*)d_in[15], (const int*)d_in[20]};
  const int*   r_rlsrc[2]   = {(const int*)d_in[16], (const int*)d_in[21]};
  const int*   r_rldst[2]   = {(const int*)d_in[17], (const int*)d_in[22]};

  const int N_L  = in_sizes[0];
  const int N_R  = in_sizes[3];
  const int M_GT = in_sizes[9] / 3;
  const int E_L  = in_sizes[11];
  const int E_RL = in_sizes[16];

  // ---- workspace layout (floats) ----
  float* w = (float*)d_ws;
  long long off = 0;
  float* acc   = w + off; off += 8;
  float* lcnt  = w + off; off += N_L;
  float* lspos = w + off; off += 3LL * N_L;
  float* lsint = w + off; off += N_L;
  float* pred[2]; float* a2[2]; float* phys[2]; float* sumd[2]; float* cntr[2];
  for (int r = 0; r < 2; ++r) {
    pred[r] = w + off; off += 3LL * N_R;
    a2[r]   = w + off; off += N_R;
    phys[r] = w + off; off += N_R;
    sumd[r] = w + off; off += N_R;
    cntr[r] = w + off; off += N_R;
  }
  const long long ws_floats = off;

  const int TB = 256;
  // 0) zero everything we accumulate into
  {
    int blocks = (int)((ws_floats + TB - 1) / TB);
    if (blocks > 4096) blocks = 4096;
    zero_kernel<<<blocks, TB, 0, stream>>>(w, ws_floats);
  }
  // 1) lidar scatter
  lidar_scatter_kernel<<<(E_L + TB - 1) / TB, TB, 0, stream>>>(
      l_src, l_dst, lidar_pos, lidar_x, lcnt, lspos, lsint, E_L);
  // 2) lidar finalize -> acc[0]
  lidar_finalize_kernel<<<(N_L + TB - 1) / TB, TB, 0, stream>>>(
      lidar_logvar, lidar_pos, lidar_x, lcnt, lspos, lsint, acc + 0, N_L);

  // radars -> acc[1]
  for (int r = 0; r < 2; ++r) {
    radar_pred_kernel<<<(N_R + TB - 1) / TB, TB, 0, stream>>>(
        r_pos[r], r_x[r], r_tsrc[r], r_tdst[r], r_batch[r], dt_sec,
        pred[r], a2[r], N_R);
    {
      int waves = (N_R + 15) / 16;
      int wavesPerBlk = 4;                 // 128 threads
      int blocks = (waves + wavesPerBlk - 1) / wavesPerBlk;
      radar_nn_wmma_kernel<<<blocks, wavesPerBlk * 32, 0, stream>>>(
          pred[r], a2[r], gt_pos, r_tsrc[r], phys[r], N_R, M_GT);
    }
    radar_rl_scatter_kernel<<<(E_RL + TB - 1) / TB, TB, 0, stream>>>(
        r_rlsrc[r], r_rldst[r], r_pos[r], lidar_pos, sumd[r], cntr[r], E_RL);
    radar_finalize_kernel<<<(N_R + TB - 1) / TB, TB, 0, stream>>>(
        r_logvar[r], r_batch[r], dt_sec, phys[r], sumd[r], cntr[r], acc + 1, N_R);
  }

  final_kernel<<<1, 32, 0, stream>>>(acc, (float*)d_out, N_L, N_R);
  (void)n_in; (void)out_size; (void)ws_size;
}